// MinEuclideanDistBlock_46480136077428
// MI455X (gfx1250) — compile-verified
//
#include <hip/hip_runtime.h>

// CDNA5 / gfx1250 shapelet min-distance kernel.
// Core op: V_WMMA_F32_16X16X4_F32 (native fp32 matrix pipe, wave32).

typedef __attribute__((ext_vector_type(2))) float v2f;
typedef __attribute__((ext_vector_type(8))) float v8f;

#define SH_S   64      // shapelet length
#define SH_K   128     // number of shapelets
#define SH_C   8       // channels
#define SH_L   2048    // series length
#define SH_W   1985    // windows = L - S + 1
#define SH_N   64      // batch
#define WTILE  64      // windows per block (4 M-tiles of 16)

// ---------------------------------------------------------------------------
// Kernel 1: z-normalize shapelets, store normalized values + sum-of-squares.
// One thread per (c,k) shapelet row of 64 contiguous floats.
// ---------------------------------------------------------------------------
__global__ void znorm_kernel(const float* __restrict__ sh,
                             float* __restrict__ shn,
                             float* __restrict__ sqs) {
    int t = blockIdx.x * blockDim.x + threadIdx.x;   // 0..C*K-1
    if (t >= SH_C * SH_K) return;
    const float* s = sh + (size_t)t * SH_S;
    float mu = 0.f;
    #pragma unroll 8
    for (int i = 0; i < SH_S; ++i) mu += s[i];
    mu *= (1.0f / SH_S);
    float var = 0.f;
    #pragma unroll 8
    for (int i = 0; i < SH_S; ++i) { float d = s[i] - mu; var += d * d; }
    var *= (1.0f / SH_S);
    float inv = rsqrtf(var);                 // matches (s-mu)/std
    float ss = 0.f;
    float* o = shn + (size_t)t * SH_S;
    #pragma unroll 8
    for (int i = 0; i < SH_S; ++i) {
        float z = (s[i] - mu) * inv;
        o[i] = z;
        ss += z * z;
    }
    sqs[t] = ss;
}

// ---------------------------------------------------------------------------
// Kernel 2: fused cross-GEMM (WMMA f32 16x16x4) + distance + sum_c + min_k.
// Block: 256 threads = 8 waves. Wave j handles shapelet k-tile j (16 k's).
// Grid: (32 window-tiles, 64 batch).
// A tile (64w x 64s) is Hankel: built from 127 contiguous floats in LDS.
// B fragments (16 x v2f per lane) are preloaded into registers before the
// WMMA loop so global-load latency is paid once per channel, not per k-step.
// ---------------------------------------------------------------------------
__global__ __launch_bounds__(256, 1)
void shapelet_wmma_kernel(const float* __restrict__ x,
                          const float* __restrict__ shn,
                          const float* __restrict__ sqs,
                          float* __restrict__ out) {
    __shared__ float xs[WTILE + SH_S];       // 128 floats: x[n,c, w0 .. w0+126]
    __shared__ float sqx[WTILE];             // sliding-window sum of squares
    __shared__ float minbuf[8][WTILE];       // per-wave k-mins

    const int n    = blockIdx.y;
    const int w0   = blockIdx.x * WTILE;
    const int tid  = threadIdx.x;
    const int wave = tid >> 5;               // k-tile index 0..7
    const int lane = tid & 31;
    const int mrow = lane & 15;              // M row (A) / N col (B,C,D)
    const int half = lane >> 4;              // 0 or 1

    // Persistent distance-sum accumulators: 4 M-tiles x v8f (w = mt*16 + r + 8*half)
    v8f dsum[4];
    #pragma unroll
    for (int mt = 0; mt < 4; ++mt) dsum[mt] = (v8f){0,0,0,0,0,0,0,0};

    for (int c = 0; c < SH_C; ++c) {
        __syncthreads();   // previous iteration's LDS consumers done
        // Stage 127 contiguous x values (clamped at series end; invalid
        // windows are masked at the final store).
        if (tid < WTILE + SH_S - 1) {
            int gi = w0 + tid;
            if (gi > SH_L - 1) gi = SH_L - 1;
            xs[tid] = x[((size_t)n * SH_C + c) * SH_L + gi];
            // Prefetch next channel's tile into near caches (global_prefetch_b8).
            if (c + 1 < SH_C)
                __builtin_prefetch(&x[((size_t)n * SH_C + (c + 1)) * SH_L + gi], 0, 3);
        }
        __syncthreads();
        // sq_x for the 64 windows of this tile.
        if (tid < WTILE) {
            float acc = 0.f;
            #pragma unroll 8
            for (int s = 0; s < SH_S; ++s) { float v = xs[tid + s]; acc += v * v; }
            sqx[tid] = acc;
        }

        // ---- Preload the wave's full B tile for this channel into registers.
        // B fragment base: shapelet (k-tile, row = mrow), s offset half*2.
        // bfrag[ks] holds B[ks*4 + half*2 + {0,1}][mrow]  (8B-aligned loads).
        const float* Bp = shn + ((size_t)(c * SH_K + wave * 16 + mrow)) * SH_S + half * 2;
        v2f bfrag[16];
        #pragma unroll
        for (int ks = 0; ks < 16; ++ks)
            bfrag[ks] = *(const v2f*)(Bp + ks * 4);
        const float myss = sqs[c * SH_K + wave * 16 + mrow];

        v8f cc[4];
        #pragma unroll
        for (int mt = 0; mt < 4; ++mt) cc[mt] = (v8f){0,0,0,0,0,0,0,0};

        __syncthreads();   // xs + sqx visible to all waves

        // 16 k-steps of 4 over the shapelet length, 4 M-tiles each.
        // All operands now come from registers (B) and LDS (A, Hankel-shared).
        #pragma unroll
        for (int ks = 0; ks < 16; ++ks) {
            #pragma unroll
            for (int mt = 0; mt < 4; ++mt) {
                // A 16x4 f32 fragment from Hankel LDS tile:
                // A[m][k] = xs[mt*16 + m + ks*4 + k], lane holds k = half*2 + {0,1}
                const int ai = mt * 16 + mrow + ks * 4 + half * 2;
                v2f a;
                a.x = xs[ai];
                a.y = xs[ai + 1];
                cc[mt] = __builtin_amdgcn_wmma_f32_16x16x4_f32(
                    /*neg_a=*/false, a, /*neg_b=*/false, bfrag[ks],
                    /*c_mod=*/(short)0, cc[mt],
                    /*reuse_a=*/false, /*reuse_b=*/false);
            }
        }

        // Fused epilogue: d = sqrt(max(sq_x + sq_s - 2*cross, 1e-12)), sum over c.
        #pragma unroll
        for (int mt = 0; mt < 4; ++mt) {
            #pragma unroll
            for (int r = 0; r < 8; ++r) {
                const int m = r + 8 * half;          // C/D layout row
                float d2 = sqx[mt * 16 + m] + myss - 2.0f * cc[mt][r];
                d2 = fmaxf(d2, 1e-12f);
                dsum[mt][r] += sqrtf(d2);
            }
        }
    }

    // Min over the wave's 16 shapelets: butterfly within 16-lane groups
    // (xor masks 8..1 never cross bit 4, so both halves reduce independently).
    #pragma unroll
    for (int mt = 0; mt < 4; ++mt) {
        #pragma unroll
        for (int r = 0; r < 8; ++r) {
            float v = dsum[mt][r];
            v = fminf(v, __shfl_xor(v, 8, 32));
            v = fminf(v, __shfl_xor(v, 4, 32));
            v = fminf(v, __shfl_xor(v, 2, 32));
            v = fminf(v, __shfl_xor(v, 1, 32));
            if (mrow == 0) minbuf[wave][mt * 16 + r + 8 * half] = v;
        }
    }
    __syncthreads();

    // Min across the 8 waves (8 k-tiles) and store valid windows.
    if (tid < WTILE) {
        float v = minbuf[0][tid];
        #pragma unroll
        for (int j = 1; j < 8; ++j) v = fminf(v, minbuf[j][tid]);
        const int w = w0 + tid;
        if (w < SH_W) out[(size_t)n * SH_W + w] = v;
    }
}

// ---------------------------------------------------------------------------
extern "C" void kernel_launch(void* const* d_in, const int* in_sizes, int n_in,
                              void* d_out, int out_size, void* d_ws, size_t ws_size,
                              hipStream_t stream) {
    const float* x  = (const float*)d_in[0];   // (64, 8, 2048) f32
    const float* sh = (const float*)d_in[1];   // (8, 128, 64)  f32
    float* out = (float*)d_out;                // (64, 1, 1985) f32

    // Workspace: normalized shapelets (C*K*S f32 = 256KB) + sq_s (C*K f32)
    float* shn = (float*)d_ws;
    float* sqs = shn + (size_t)SH_C * SH_K * SH_S;

    znorm_kernel<<<(SH_C * SH_K + 255) / 256, 256, 0, stream>>>(sh, shn, sqs);

    dim3 grid((SH_W + WTILE - 1) / WTILE, SH_N);   // (32, 64)
    shapelet_wmma_kernel<<<grid, 256, 0, stream>>>(x, shn, sqs, out);
}